// TGCN_11433202942308
// MI455X (gfx1250) — compile-verified
//
#include <hip/hip_runtime.h>

// ---------------------------------------------------------------------------
// TGCN forward for MI455X (gfx1250, wave32). All GEMMs use
// V_WMMA_F32_16X16X4_F32 (exact fp32 FMA on the matrix pipe).
// Conv inputs use time-padded rows (stride 514, zero cols at 0 and 513) so the
// 3-tap temporal conv inner loop is branch-free.
// ---------------------------------------------------------------------------

typedef __attribute__((ext_vector_type(2))) float v2f;
typedef __attribute__((ext_vector_type(8))) float v8f;

#define TP 514  // padded time stride

__device__ inline v8f wmma4(v2f a, v2f b, v8f c) {
  // 8 args: (neg_a, A, neg_b, B, c_mod, C, reuse_a, reuse_b)
  return __builtin_amdgcn_wmma_f32_16x16x4_f32(false, a, false, b, (short)0, c,
                                               false, false);
}

__device__ inline v2f ld2(const float* p) { return *(const v2f*)p; }

__device__ inline float leaky(float x) { return x > 0.0f ? x : 0.01f * x; }

// Skeleton adjacency (compile-time constant from the reference)
__device__ __constant__ int NB_IDX[19 * 9] = {
    1, 2, 3, 4, 0, 0, 0, 0, 0,  //
    0, 4, 5, 6, 0, 0, 0, 0, 0,  //
    0, 3, 4, 6, 7, 8, 0, 0, 0,  //
    0, 2, 4, 5, 8, 9, 0, 0, 0,  //
    0, 1, 3, 5, 9, 0, 0, 0, 0,  //
    1, 3, 4, 6, 9, 10, 0, 0, 0,  //
    1, 2, 4, 5, 10, 11, 0, 0, 0,  //
    2, 8, 11, 12, 13, 0, 0, 0, 0,  //
    2, 3, 4, 7, 9, 10, 12, 13, 14,  //
    3, 4, 5, 8, 10, 13, 14, 15, 0,  //
    4, 5, 6, 8, 9, 11, 14, 15, 16,  //
    6, 7, 10, 15, 16, 0, 0, 0, 0,  //
    7, 8, 13, 16, 17, 0, 0, 0, 0,  //
    7, 8, 9, 12, 14, 15, 17, 0, 0,  //
    8, 9, 10, 13, 15, 17, 18, 0, 0,  //
    9, 10, 11, 13, 14, 16, 18, 0, 0,  //
    10, 11, 12, 15, 18, 0, 0, 0, 0,  //
    12, 13, 14, 18, 0, 0, 0, 0, 0,  //
    14, 15, 16, 17, 0, 0, 0, 0, 0};
__device__ __constant__ int NB_DEG[19] = {4, 4, 6, 6, 5, 6, 6, 5, 9, 8,
                                          9, 5, 5, 7, 7, 7, 5, 4, 4};

// ---------------------------------------------------------------------------
// Kernel 0: x (4,2,19,512,200) -> h0[bn][node][f][pad | t | pad] (stride 514)
// ---------------------------------------------------------------------------
__global__ void k_transpose_in(const float* __restrict__ x,
                               float* __restrict__ h) {
  int idx = blockIdx.x * blockDim.x + threadIdx.x;
  if (idx >= 8 * 19 * 200 * 512) return;
  int t = idx & 511;
  int rest = idx >> 9;  // (bn*19+node)*200 + f
  float* row = h + (size_t)rest * TP;
  row[1 + t] = x[(size_t)(((rest / 200) * 512 + t) * 200) + (rest % 200)];
  // rest/200 == bn*19+node ; x index = ((bn*19+node)*512+t)*200 + f
  if (t == 0) row[0] = 0.0f;
  if (t == 511) row[513] = 0.0f;
}

// ---------------------------------------------------------------------------
// Kernel R: repack w1 (Fout,Fin,3) -> w1t[tap][fout][fin] (contiguous A rows)
// ---------------------------------------------------------------------------
__global__ void k_repack_w1(const float* __restrict__ w1,
                            float* __restrict__ w1t, int Fin, int Fout) {
  int idx = blockIdx.x * blockDim.x + threadIdx.x;
  if (idx >= 3 * Fout * Fin) return;
  int fin = idx % Fin;
  int rest = idx / Fin;
  int fo = rest % Fout;
  int tap = rest / Fout;
  w1t[idx] = w1[((size_t)fo * Fin + fin) * 3 + tap];
}

// ---------------------------------------------------------------------------
// Kernel 1: temporal conv (k=3, pad 1) + fused BN affine.
// Input padded (stride TP); output h2 packed (stride 512).
// grid = (2, Fout/16, 152), block = 128 (4 waves, each 16x64 of output).
// ---------------------------------------------------------------------------
__global__ void k_conv_bn(const float* __restrict__ hin,
                          const float* __restrict__ w1t,
                          const float* __restrict__ b1,
                          const float* __restrict__ gam,
                          const float* __restrict__ bet,
                          const float* __restrict__ rm,
                          const float* __restrict__ rv,
                          float* __restrict__ h2, int Fin, int Fout) {
  const int lane = threadIdx.x & 31;
  const int wave = threadIdx.x >> 5;
  const int m = lane & 15;     // A row / B col / D col
  const int ksel = lane >> 4;  // selects K pair {0,1} vs {2,3}
  const int m0 = blockIdx.y << 4;
  const int bnode = blockIdx.z;
  const int tcol = (blockIdx.x << 8) + (wave << 6) + m;
  const int PS = Fout * Fin;  // w1t tap-plane stride

  const float* __restrict__ xin = hin + (size_t)bnode * Fin * TP;

  v8f acc[4] = {v8f{}, v8f{}, v8f{}, v8f{}};

  for (int kk = 0; kk < Fin; kk += 4) {
    const int f0 = kk + (ksel << 1);
    const float* __restrict__ row1 = xin + (size_t)f0 * TP + 1;  // skip pad
    const float* __restrict__ wbase = w1t + (size_t)(m0 + m) * Fin + f0;
    v2f a0 = ld2(wbase);
    v2f a1 = ld2(wbase + PS);
    v2f a2 = ld2(wbase + 2 * PS);
    __builtin_prefetch(row1 + 4 * TP + tcol, 0, 0);  // next K-chunk B row
#pragma unroll
    for (int j = 0; j < 4; ++j) {
      const int t = tcol + (j << 4);
      v2f b;
      b.x = row1[t - 1];
      b.y = row1[TP + t - 1];
      acc[j] = wmma4(a0, b, acc[j]);
      b.x = row1[t];
      b.y = row1[TP + t];
      acc[j] = wmma4(a1, b, acc[j]);
      b.x = row1[t + 1];
      b.y = row1[TP + t + 1];
      acc[j] = wmma4(a2, b, acc[j]);
    }
  }

  // BN folded: y = conv*scale + ((b1 - rm)*scale + beta)
#pragma unroll
  for (int j = 0; j < 8; ++j) {
    const int fo = m0 + j + (ksel << 3);
    const float sc = gam[fo] * rsqrtf(rv[fo] + 1e-5f);
    const float off = (b1[fo] - rm[fo]) * sc + bet[fo];
    float* __restrict__ orow = h2 + ((size_t)bnode * Fout + fo) * 512;
#pragma unroll
    for (int l = 0; l < 4; ++l) orow[tcol + (l << 4)] = acc[l][j] * sc + off;
  }
}

// ---------------------------------------------------------------------------
// Kernel 2: z[node][f][t] = leaky( max over neighbors of h2[nb][f][t] )
// (both packed, stride 512)
// ---------------------------------------------------------------------------
__global__ void k_nbmax(const float* __restrict__ h2, float* __restrict__ z,
                        int Fout, int total) {
  int idx = blockIdx.x * blockDim.x + threadIdx.x;
  if (idx >= total) return;
  int t = idx & 511;
  int rest = idx >> 9;
  int f = rest % Fout;
  rest /= Fout;
  int node = rest % 19;
  int bn = rest / 19;
  const int deg = NB_DEG[node];
  float v = -3.4e38f;
  for (int i = 0; i < deg; ++i) {
    int nb = NB_IDX[node * 9 + i];
    v = fmaxf(v, h2[((size_t)(bn * 19 + nb) * Fout + f) * 512 + t]);
  }
  z[idx] = leaky(v);
}

// ---------------------------------------------------------------------------
// Kernel 3: out = leaky( W2 (Fout x 2Fout) * hc (2Fout x 512) + b2 )
// hc rows: c < Fout -> leaky(h2 row c); c >= Fout -> z row (leaky pre-applied)
// Output stride `ots` (514 padded for next conv; 512 packed for final layer).
// ---------------------------------------------------------------------------
__global__ void k_w2(const float* __restrict__ h2, const float* __restrict__ z,
                     const float* __restrict__ w2,
                     const float* __restrict__ b2, float* __restrict__ out,
                     int Fout, int ots) {
  const int K2 = Fout * 2;
  const int lane = threadIdx.x & 31;
  const int wave = threadIdx.x >> 5;
  const int m = lane & 15;
  const int ksel = lane >> 4;
  const int m0 = blockIdx.y << 4;
  const int bnode = blockIdx.z;
  const int tcol = (blockIdx.x << 8) + (wave << 6) + m;

  const float* __restrict__ h2b = h2 + (size_t)bnode * Fout * 512;
  const float* __restrict__ zb = z + (size_t)bnode * Fout * 512;

  v8f acc[4] = {v8f{}, v8f{}, v8f{}, v8f{}};

  for (int kk = 0; kk < K2; kk += 4) {
    const int c = kk + (ksel << 1);
    v2f a = ld2(w2 + (size_t)(m0 + m) * K2 + c);
    const bool isH = (c < Fout);  // uniform across the wave (Fout % 4 == 0)
    const float* __restrict__ row =
        isH ? (h2b + (size_t)c * 512) : (zb + (size_t)(c - Fout) * 512);
#pragma unroll
    for (int j = 0; j < 4; ++j) {
      const int t = tcol + (j << 4);
      float bx = row[t];
      float by = row[512 + t];
      if (isH) {
        bx = leaky(bx);
        by = leaky(by);
      }
      v2f b;
      b.x = bx;
      b.y = by;
      acc[j] = wmma4(a, b, acc[j]);
    }
  }

  const int off = (ots == TP) ? 1 : 0;  // skip left pad when padded
#pragma unroll
  for (int j = 0; j < 8; ++j) {
    const int fo = m0 + j + (ksel << 3);
    const float bias = b2[fo];
    float* __restrict__ orow = out + ((size_t)bnode * Fout + fo) * ots;
#pragma unroll
    for (int l = 0; l < 4; ++l) {
      const int t = tcol + (l << 4);
      orow[off + t] = leaky(acc[l][j] + bias);
      if (off) {
        if (t == 0) orow[0] = 0.0f;
        if (t == 511) orow[513] = 0.0f;
      }
    }
  }
}

// ---------------------------------------------------------------------------
// Kernel 4: hm[(bn*512+t)*256 + f] = mean over 19 nodes of h[bn][node][f][t]
// (final h is packed, stride 512)
// ---------------------------------------------------------------------------
__global__ void k_mean(const float* __restrict__ h, float* __restrict__ hm) {
  int idx = blockIdx.x * blockDim.x + threadIdx.x;
  if (idx >= 8 * 512 * 256) return;
  int f = idx & 255;
  int t = (idx >> 8) & 511;
  int bn = idx >> 17;
  float s = 0.0f;
  for (int node = 0; node < 19; ++node)
    s += h[((size_t)(bn * 19 + node) * 256 + f) * 512 + t];
  hm[idx] = s * (1.0f / 19.0f);
}

// ---------------------------------------------------------------------------
// Kernel 5: out(R x Fo) = X(R x Fi) * W^T + bias   (row-major, R = 4096)
// grid = (R/64, ceil(Fo/16)), block = 128 (waves stacked in M).
// ---------------------------------------------------------------------------
__global__ void k_linear(const float* __restrict__ X,
                         const float* __restrict__ W,
                         const float* __restrict__ bias,
                         float* __restrict__ out, int Fi, int Fo) {
  const int lane = threadIdx.x & 31;
  const int wave = threadIdx.x >> 5;
  const int m = lane & 15;
  const int ksel = lane >> 4;
  const int r0 = (blockIdx.x << 6) + (wave << 4);
  const int n0 = blockIdx.y << 4;
  const int fo = n0 + m;  // B column for this lane

  v8f acc = v8f{};
  for (int kk = 0; kk < Fi; kk += 4) {
    const int c = kk + (ksel << 1);
    v2f a = ld2(X + (size_t)(r0 + m) * Fi + c);
    v2f b;
    if (fo < Fo) {
      b = ld2(W + (size_t)fo * Fi + c);
    } else {
      b.x = 0.0f;
      b.y = 0.0f;
    }
    acc = wmma4(a, b, acc);
  }
#pragma unroll
  for (int j = 0; j < 8; ++j) {
    const int r = r0 + j + (ksel << 3);
    if (fo < Fo) out[(size_t)r * Fo + fo] = acc[j] + bias[fo];
  }
}

// ---------------------------------------------------------------------------
extern "C" void kernel_launch(void* const* d_in, const int* in_sizes, int n_in,
                              void* d_out, int out_size, void* d_ws,
                              size_t ws_size, hipStream_t stream) {
  (void)in_sizes;
  (void)n_in;
  (void)out_size;
  (void)ws_size;

  const float* x = (const float*)d_in[0];
  float* ws = (float*)d_ws;
  const size_t HP = (size_t)8 * 19 * 256 * TP;  // 20,000,768 padded floats
  const size_t H = 19922944;                    // 8*19*256*512 packed floats
  float* bufA = ws;            // padded ping
  float* bufB = ws + HP;       // padded pong
  float* bufC = ws + 2 * HP;   // packed h2 scratch
  float* w1t = ws + 2 * HP + H;  // repacked conv weights (<= 196,608)

  // Input transpose -> padded h0 in bufA
  {
    const int total = 8 * 19 * 200 * 512;
    k_transpose_in<<<(total + 255) / 256, 256, 0, stream>>>(x, bufA);
  }

  static const int FIN[8] = {200, 32, 32, 64, 64, 128, 128, 256};
  static const int FOUT[8] = {32, 32, 64, 64, 128, 128, 256, 256};

  float* cur = bufA;
  float* nxt = bufB;
  float* h2 = bufC;
  for (int L = 0; L < 8; ++L) {
    const float* w1p = (const float*)d_in[1 + L * 8 + 0];
    const float* b1p = (const float*)d_in[1 + L * 8 + 1];
    const float* gap = (const float*)d_in[1 + L * 8 + 2];
    const float* bep = (const float*)d_in[1 + L * 8 + 3];
    const float* rmp = (const float*)d_in[1 + L * 8 + 4];
    const float* rvp = (const float*)d_in[1 + L * 8 + 5];
    const float* w2p = (const float*)d_in[1 + L * 8 + 6];
    const float* b2p = (const float*)d_in[1 + L * 8 + 7];

    const int wtot = 3 * FOUT[L] * FIN[L];
    k_repack_w1<<<(wtot + 255) / 256, 256, 0, stream>>>(w1p, w1t, FIN[L],
                                                        FOUT[L]);
    dim3 g(2, FOUT[L] / 16, 152);
    k_conv_bn<<<g, 128, 0, stream>>>(cur, w1t, b1p, gap, bep, rmp, rvp, h2,
                                     FIN[L], FOUT[L]);
    const int totalz = 152 * FOUT[L] * 512;
    // z overwrites 'cur' (conv input no longer needed); packed layout
    k_nbmax<<<(totalz + 255) / 256, 256, 0, stream>>>(h2, cur, FOUT[L], totalz);
    const int ots = (L == 7) ? 512 : TP;  // last layer packed for h_flat
    k_w2<<<g, 128, 0, stream>>>(h2, cur, w2p, b2p, nxt, FOUT[L], ots);
    float* tmp = cur;
    cur = nxt;
    nxt = tmp;
  }
  // After 8 layers cur == bufA holds final h packed (F=256). bufC is free.

  float* hm = bufC;              // 4096 x 256
  float* l0 = bufC + (1 << 20);  // 4096 x 512
  float* l1 = l0 + 4096 * 512;   // 4096 x 512

  k_mean<<<(8 * 512 * 256 + 255) / 256, 256, 0, stream>>>(cur, hm);

  const float* W0 = (const float*)d_in[65];
  const float* B0 = (const float*)d_in[66];
  const float* W1 = (const float*)d_in[67];
  const float* B1 = (const float*)d_in[68];
  const float* W2 = (const float*)d_in[69];
  const float* B2 = (const float*)d_in[70];

  k_linear<<<dim3(64, 32), 128, 0, stream>>>(hm, W0, B0, l0, 256, 512);
  k_linear<<<dim3(64, 32), 128, 0, stream>>>(l0, W1, B1, l1, 512, 512);
  // Last linear (Fo=2) writes y straight into d_out[0:8192)
  k_linear<<<dim3(64, 1), 128, 0, stream>>>(l1, W2, B2, (float*)d_out, 512, 2);

  // h_flat = reshape only: flat layouts match -> pure d2d copy
  hipMemcpyAsync((float*)d_out + 8192, cur, H * sizeof(float),
                 hipMemcpyDeviceToDevice, stream);
}